// Complex_net_ext_87058987090499
// MI455X (gfx1250) — compile-verified
//
#include <hip/hip_runtime.h>

// ---------------------------------------------------------------------------
// Reduced computation (see analysis):
//   Stage 1: u[b,n] = sum_c xr[b,n,c]*W1r[63,c] - xi[b,n,c]*W1i[63,c]   (n>=1)
//            v[b,n] = sum_c xi[b,n,c]*W1r[63,c] + xr[b,n,c]*W1i[63,c]   (n>=1)
//            u[b,0] = xr[b,0,63], v[b,0] = xi[b,0,63]
//   Stage 2: out[b,m] = | sum_n u[b,n]*(W2r+W2i)[m,n] + v[b,n]*(W2r-W2i)[m,n] |
// Stage 1 streams the full 268 MB input once -> memory bound (~11.5 us @ 23.3 TB/s).
// Stage 2 is a (8192x64)@(64x64) f32 GEMM pair -> V_WMMA_F32_16X16X4_F32.
// ---------------------------------------------------------------------------

typedef float v2f __attribute__((ext_vector_type(2)));
typedef float v4f __attribute__((ext_vector_type(4)));
typedef float v8f __attribute__((ext_vector_type(8)));

#define NB 8192
#define NN 64

// ---------------------------------------------------------------------------
// Stage 1: one wave32 per (b,n) row. Each lane loads one float4 (lanes cover
// the whole 512-byte row in a single perfectly-coalesced load), forms partial
// complex dot products, then a 5-step shfl_xor wave reduction.
// Non-temporal loads: x is streamed exactly once, no reuse to cache.
// ---------------------------------------------------------------------------
__global__ __launch_bounds__(256) void stage1_uv(
    const float* __restrict__ x,
    const float* __restrict__ W1r,
    const float* __restrict__ W1i,
    float* __restrict__ U,
    float* __restrict__ V)
{
    const int lane   = threadIdx.x & 31;
    const int wave   = (blockIdx.x * blockDim.x + threadIdx.x) >> 5;
    const int nwaves = (gridDim.x * blockDim.x) >> 5;

    // Per-lane weights: lane l owns complex columns c = 2l, 2l+1 of row 63.
    const float wr0 = W1r[63 * NN + 2 * lane];
    const float wr1 = W1r[63 * NN + 2 * lane + 1];
    const float wi0 = W1i[63 * NN + 2 * lane];
    const float wi1 = W1i[63 * NN + 2 * lane + 1];

    const int total_rows = NB * NN;
    for (int row = wave; row < total_rows; row += nwaves) {
        const v4f* rp = (const v4f*)(x + (size_t)row * 128 + 4 * lane);
        v4f z = __builtin_nontemporal_load(rp);   // (re[2l], im[2l], re[2l+1], im[2l+1])

        float u = z.x * wr0 - z.y * wi0 + z.z * wr1 - z.w * wi1;
        float v = z.x * wi0 + z.y * wr0 + z.z * wi1 + z.w * wr1;

        // Row n==0 passthrough values live in lane 31's float4 (c = 62, 63).
        const float re63 = __shfl(z.z, 31, 32);
        const float im63 = __shfl(z.w, 31, 32);

        #pragma unroll
        for (int m = 16; m >= 1; m >>= 1) {
            u += __shfl_xor(u, m, 32);
            v += __shfl_xor(v, m, 32);
        }

        if ((row & (NN - 1)) == 0) { u = re63; v = im63; }

        if (lane == 0) {
            U[row] = u;
            V[row] = v;
        }
    }
}

// ---------------------------------------------------------------------------
// Stage 2: out = | U @ (W2r+W2i)^T + V @ (W2r-W2i)^T | via V_WMMA_F32_16X16X4_F32.
// Tile: M = 16 batches, N = 16 output rows m, K = n stepped by 4.
// ISA operand layouts (CDNA5 ISA):
//   A 16x4 f32 : lane -> M = lane&15, VGPR j holds K = (lane>>4)*2 + j
//   B 4x16 f32 : lane -> N = lane&15, VGPR j holds K = (lane>>4)*2 + j
// => both operands are a single aligned float2 load of K-consecutive elements
//    from row-major U/V and row-major weights. 32 WMMAs per tile.
// One wave per tile; 2048 tiles = 256 blocks x 8 waves, EXEC all-ones throughout.
// ---------------------------------------------------------------------------
__global__ __launch_bounds__(256) void stage2_wmma(
    const float* __restrict__ U,
    const float* __restrict__ V,
    const float* __restrict__ W2r,
    const float* __restrict__ W2i,
    float* __restrict__ out)
{
    const int lane = threadIdx.x & 31;
    const int wave = (blockIdx.x * blockDim.x + threadIdx.x) >> 5;

    const int b0 = (wave >> 2) << 4;   // 512 batch tiles of 16
    const int m0 = (wave & 3) << 4;    // 4 m tiles of 16
    const int lm = lane & 15;
    const int lk = (lane >> 4) << 1;   // 0 or 2

    const float* urow = U   + (size_t)(b0 + lm) * NN;
    const float* vrow = V   + (size_t)(b0 + lm) * NN;
    const float* wr   = W2r + (size_t)(m0 + lm) * NN;
    const float* wi   = W2i + (size_t)(m0 + lm) * NN;

    v8f acc = {};
    #pragma unroll
    for (int k0 = 0; k0 < NN; k0 += 4) {
        v2f au = *(const v2f*)(urow + k0 + lk);
        v2f av = *(const v2f*)(vrow + k0 + lk);
        v2f r2 = *(const v2f*)(wr   + k0 + lk);
        v2f i2 = *(const v2f*)(wi   + k0 + lk);
        v2f ba = r2 + i2;   // pairs with U
        v2f bb = r2 - i2;   // pairs with V
        // 8 args: (neg_a, A, neg_b, B, c_mod, C, reuse_a, reuse_b)
        acc = __builtin_amdgcn_wmma_f32_16x16x4_f32(false, au, false, ba,
                                                    (short)0, acc, false, false);
        acc = __builtin_amdgcn_wmma_f32_16x16x4_f32(false, av, false, bb,
                                                    (short)0, acc, false, false);
    }

    // C/D layout: VGPR r -> M = r + (lane>=16 ? 8 : 0), N = lane&15
    const int rbase = b0 + ((lane >> 4) << 3);
    #pragma unroll
    for (int r = 0; r < 8; ++r) {
        out[(size_t)(rbase + r) * NN + m0 + lm] = fabsf(acc[r]);
    }
}

extern "C" void kernel_launch(void* const* d_in, const int* in_sizes, int n_in,
                              void* d_out, int out_size, void* d_ws, size_t ws_size,
                              hipStream_t stream) {
    const float* x   = (const float*)d_in[0];
    const float* W1r = (const float*)d_in[1];
    const float* W1i = (const float*)d_in[2];
    const float* W2r = (const float*)d_in[3];
    const float* W2i = (const float*)d_in[4];
    float* out = (float*)d_out;

    float* U = (float*)d_ws;            // NB*NN floats (2 MB)
    float* V = U + (size_t)NB * NN;     // NB*NN floats (2 MB)

    // Stage 1: 1024 blocks x 256 threads = 8192 waves, grid-stride over 524288 rows.
    stage1_uv<<<1024, 256, 0, stream>>>(x, W1r, W1i, U, V);

    // Stage 2: exactly 2048 tiles = 256 blocks x 8 waves.
    stage2_wmma<<<256, 256, 0, stream>>>(U, V, W2r, W2i, out);
}